// ForgetMult_1580547972331
// MI455X (gfx1250) — compile-verified
//
#include <hip/hip_runtime.h>
#include <hip/hip_bf16.h>
#include <stdint.h>

// Problem geometry (matches reference)
#define SEQ   2048
#define BATCH 32
#define DIM   1024
#define C     (BATCH * DIM)        // 32768 independent channels

// Kernel geometry: 1 wave per block so the wave owns its TENSORcnt.
#define BLOCK 32                   // threads per block (exactly 1 wave32)
#define VEC   4                    // channels per thread (float4)
#define CHB   (BLOCK * VEC)        // 128 channels per block
#define GRID  (C / CHB)            // 256 blocks
#define TT    16                   // timesteps per TDM tile
#define NT    (SEQ / TT)           // 128 tiles
#define NBUF  8                    // LDS ring depth (prefetch distance)
#define SLOT  (TT * CHB)           // floats per ring slot per array (2048 = 8 KB)
#define SMEM_FLOATS (2 * NBUF * SLOT)          // 32768 floats = 128 KB
#define SMEM_BYTES  (SMEM_FLOATS * 4)
// 2 tensor ops per tile (f + x); oldest tile done when <= (NBUF-1)*2 outstanding.
#define TENSOR_WAIT ((NBUF - 1) * 2)           // 14

typedef uint32_t v4u __attribute__((ext_vector_type(4)));
typedef uint32_t v8u __attribute__((ext_vector_type(8)));

__global__ __launch_bounds__(BLOCK)
void forget_mult_tdm(const float* __restrict__ f,
                     const float* __restrict__ x,
                     const float* __restrict__ h0,
                     float* __restrict__ out) {
    extern __shared__ float smem[];
    float* sf = smem;                  // [NBUF][TT][CHB]
    float* sx = smem + NBUF * SLOT;    // [NBUF][TT][CHB]

    const int tid     = threadIdx.x;
    const int chBlock = blockIdx.x * CHB;       // first channel of this block
    const int ch0     = chBlock + tid * VEC;    // first channel of this thread

    // ---- Tensor DMA descriptor group 1 (loop-invariant), per ISA 08 §8.4 ----
    // dword0: workgroup_mask=0 (no cluster), data_size=2 (4 B), no pad/iterate.
    // tensor_dim0 = tile_dim0 = CHB, tensor_dim1 = tile_dim1 = TT (no OOB),
    // tensor_dim0_stride = C elements between timesteps.
    v8u g1;
    g1[0] = 0x00020000u;                       // data_size = 4 bytes
    g1[1] = ((uint32_t)CHB & 0xFFFFu) << 16;   // tensor_dim0[15:0]
    g1[2] = ((uint32_t)TT) << 16;              // tensor_dim0[31:16]=0 | tensor_dim1[15:0]
    g1[3] = ((uint32_t)CHB) << 16;             // tensor_dim1[31:16]=0 | tile_dim0
    g1[4] = (uint32_t)TT;                      // tile_dim1 | tile_dim2=0
    g1[5] = (uint32_t)C;                       // tensor_dim0_stride[31:0]
    g1[6] = 0u;                                // stride0[47:32]=0 | stride1[15:0]=0
    g1[7] = 0u;                                // stride1[47:16]=0

    // Issue both TDM tile loads (f and x) for `tile` into ring slot tile%NBUF.
    auto issue_tile = [&](int tile) {
        const int b = tile & (NBUF - 1);
        const uint64_t elt = (uint64_t)tile * TT * C + (uint64_t)chBlock;
        const uint64_t gf  = (uint64_t)(uintptr_t)f + elt * 4u;
        const uint64_t gx  = (uint64_t)(uintptr_t)x + elt * 4u;
        // Low 32 bits of the flat shared pointer == LDS byte address.
        const uint32_t lf = (uint32_t)(uintptr_t)&sf[b * SLOT];
        const uint32_t lx = (uint32_t)(uintptr_t)&sx[b * SLOT];
        // Group 0 (ISA 08 §8.3): count=1 | lds_addr | global_addr | type=2<<30.
        v4u g0f = { 1u, lf, (uint32_t)gf, (uint32_t)(gf >> 32) | 0x80000000u };
        v4u g0x = { 1u, lx, (uint32_t)gx, (uint32_t)(gx >> 32) | 0x80000000u };
        asm volatile("tensor_load_to_lds %0, %1" :: "s"(g0f), "s"(g1) : "memory");
        asm volatile("tensor_load_to_lds %0, %1" :: "s"(g0x), "s"(g1) : "memory");
    };

    // Running hidden state (4 channels) lives in registers for all 2048 steps.
    float4 h = *(const float4*)(h0 + ch0);

    // Prime the ring: 8 tiles * 2 ops = 16 outstanding tensor ops.
#pragma unroll
    for (int t = 0; t < NBUF; ++t) issue_tile(t);

    for (int tile = 0; tile < NT; ++tile) {
        // Tensor ops complete in order within a wave: oldest tile is resident.
        __builtin_amdgcn_s_wait_tensorcnt(TENSOR_WAIT);

        const int b = tile & (NBUF - 1);
#pragma unroll
        for (int k = 0; k < TT; ++k) {
            const float4 ff = *(const float4*)&sf[(b * TT + k) * CHB + tid * VEC];
            const float4 xx = *(const float4*)&sx[(b * TT + k) * CHB + tid * VEC];
            // h = f*x + (1-f)*h  ==  h + f*(x - h)
            h.x = fmaf(ff.x, xx.x - h.x, h.x);
            h.y = fmaf(ff.y, xx.y - h.y, h.y);
            h.z = fmaf(ff.z, xx.z - h.z, h.z);
            h.w = fmaf(ff.w, xx.w - h.w, h.w);
            *(float4*)(out + (size_t)(tile * TT + k) * C + ch0) = h;
        }

        // TDM LDS writes are unordered vs this wave's DS reads: drain DS reads
        // before the DMA engine may overwrite this ring slot.
        asm volatile("s_wait_dscnt 0" ::: "memory");

        const int nxt = tile + NBUF;
        if (nxt < NT) issue_tile(nxt);
    }
}

extern "C" void kernel_launch(void* const* d_in, const int* in_sizes, int n_in,
                              void* d_out, int out_size, void* d_ws, size_t ws_size,
                              hipStream_t stream) {
    (void)in_sizes; (void)n_in; (void)out_size; (void)d_ws; (void)ws_size;
    const float* f  = (const float*)d_in[0];
    const float* x  = (const float*)d_in[1];
    const float* h0 = (const float*)d_in[2];
    float* out = (float*)d_out;
    forget_mult_tdm<<<GRID, BLOCK, SMEM_BYTES, stream>>>(f, x, h0, out);
}